// KnnFeature_28475633173115
// MI455X (gfx1250) — compile-verified
//
#include <hip/hip_runtime.h>
#include <math.h>

#define B_     4
#define DIN    64
#define NPTS   4096
#define KNN    20
#define DOUT   128
#define EPSBN  1e-5f
#define SLOPE  0.2f
#define NK     (NPTS*KNN)          // 81920 per (b, channel)

typedef float v2f __attribute__((ext_vector_type(2)));
typedef float v8f __attribute__((ext_vector_type(8)));

// ---------------------------------------------------------------- transpose
// features (B, 64, 4096) -> ptsT (B*4096, 64) row-major
__global__ void k_transpose(const float* __restrict__ feat, float* __restrict__ ptsT) {
    __shared__ float tile[32][33];
    const int b  = blockIdx.z;
    const int n0 = blockIdx.x * 32;
    const int d0 = blockIdx.y * 32;
    const int tx = threadIdx.x, ty = threadIdx.y;   // (32, 8)
#pragma unroll
    for (int i = 0; i < 4; ++i) {
        int d = d0 + ty + i * 8;
        tile[ty + i * 8][tx] = feat[((size_t)(b * DIN + d)) * NPTS + n0 + tx];
    }
    __syncthreads();
#pragma unroll
    for (int i = 0; i < 4; ++i) {
        int n = n0 + ty + i * 8;
        ptsT[((size_t)(b * NPTS + n)) * DIN + d0 + tx] = tile[tx][ty + i * 8];
    }
}

// ---------------------------------------------------------------- weight fold
// Wc[o][d] = W[o][d] - W[o][128+d]; Wn[o][d] = W[o][64+d] + W[o][128+d];
// S[o] = sum_d W[o][192+d]; also zero BN stats accumulators.
__global__ void k_prep(const float* __restrict__ W, float* __restrict__ Wc,
                       float* __restrict__ Wn, float* __restrict__ Sv,
                       float* __restrict__ stats) {
    const int o = blockIdx.x;     // 128
    const int d = threadIdx.x;    // 64
    const float* w = W + o * 256;
    Wc[o * DIN + d] = w[d]      - w[128 + d];
    Wn[o * DIN + d] = w[64 + d] + w[128 + d];
    __shared__ float red[64];
    red[d] = w[192 + d];
    __syncthreads();
    for (int s = 32; s > 0; s >>= 1) {
        if (d < s) red[d] += red[d + s];
        __syncthreads();
    }
    if (d == 0) Sv[o] = red[0];
    if (o == 0) { for (int t = d; t < 2 * DOUT; t += 64) stats[t] = 0.f; }
}

// ---------------------------------------------------------------- knn (WMMA f32)
// One wave per 16-row tile: pd = 2*P_i.P_j^T - |xi|^2 - |xj|^2, top-20 per row.
__global__ void __launch_bounds__(32) k_knn(const float* __restrict__ ptsT,
                                            int* __restrict__ knn_idx,
                                            float* __restrict__ knn_dist) {
    const int l    = threadIdx.x;
    const int b    = blockIdx.x >> 8;
    const int i0   = (blockIdx.x & 255) << 4;
    const int m    = l & 15;
    const int hi   = l >> 4;
    const int koff = hi * 2;

    __shared__ float pdt[16 * 16];
    __shared__ float mD[16][2][KNN];
    __shared__ int   mI[16][2][KNN];

    // A tile: rows i0..i0+15 (16x64), resident in VGPRs across all j tiles
    v2f a[16];
    const float* arow = ptsT + ((size_t)(b * NPTS + i0 + m)) * DIN;
    float pa = 0.f;
#pragma unroll
    for (int c = 0; c < 16; ++c) {
        a[c].x = arow[4 * c + koff];
        a[c].y = arow[4 * c + koff + 1];
        pa += a[c].x * a[c].x + a[c].y * a[c].y;
    }
    const float sqA = pa + __shfl_xor(pa, 16, 32);   // |x_{i0+m}|^2

    float dlist[KNN]; int ilist[KNN];
#pragma unroll
    for (int t = 0; t < KNN; ++t) { dlist[t] = -1e30f; ilist[t] = 0; }

    for (int jt = 0; jt < NPTS / 16; ++jt) {
        const int j0 = jt << 4;
        const float* brow = ptsT + ((size_t)(b * NPTS + j0 + m)) * DIN;
        v2f bb[16]; float pb = 0.f;
#pragma unroll
        for (int c = 0; c < 16; ++c) {
            bb[c].x = brow[4 * c + koff];
            bb[c].y = brow[4 * c + koff + 1];
            pb += bb[c].x * bb[c].x + bb[c].y * bb[c].y;
        }
        const float sqB = pb + __shfl_xor(pb, 16, 32);   // |x_{j0+m}|^2 (m == N lane)

        v8f acc = {};
#pragma unroll
        for (int c = 0; c < 16; ++c)
            acc = __builtin_amdgcn_wmma_f32_16x16x4_f32(false, a[c], false, bb[c],
                                                        (short)0, acc, false, false);
        __syncthreads();
#pragma unroll
        for (int v = 0; v < 8; ++v) {
            float sqi = __shfl(sqA, v + 8 * hi, 32);     // |x| for row M = v+8*hi
            pdt[(v + 8 * hi) * 16 + m] = 2.f * acc[v] - sqi - sqB;
        }
        __syncthreads();

        // lane handles row m, columns of parity hi
#pragma unroll
        for (int t = 0; t < 8; ++t) {
            const int col = 2 * t + hi;
            const float v = pdt[m * 16 + col];
            const int  id = j0 + col;
            if (v > dlist[KNN - 1]) {
#pragma unroll
                for (int u = KNN - 1; u >= 0; --u) {
                    if (v > dlist[u]) {
                        if (u < KNN - 1) { dlist[u + 1] = dlist[u]; ilist[u + 1] = ilist[u]; }
                        dlist[u] = v; ilist[u] = id;
                    }
                }
            }
        }
        __syncthreads();
    }

#pragma unroll
    for (int t = 0; t < KNN; ++t) { mD[m][hi][t] = dlist[t]; mI[m][hi][t] = ilist[t]; }
    __syncthreads();
    if (l < 16) {                      // merge the two sorted 20-lists per row
        int p0 = 0, p1 = 0;
        const size_t gbase = ((size_t)(b * NPTS) + i0 + l) * KNN;
#pragma unroll
        for (int t = 0; t < KNN; ++t) {
            float d0 = mD[l][0][p0], d1 = mD[l][1][p1];
            bool  ta = d0 >= d1;
            knn_dist[gbase + t] = ta ? d0 : d1;
            knn_idx [gbase + t] = ta ? mI[l][0][p0] : mI[l][1][p1];
            p0 += ta ? 1 : 0; p1 += ta ? 0 : 1;
        }
    }
}

// ---------------------------------------------------------------- cproj GEMM
// cproj (B*N, 128) = ptsT (B*N, 64) x Wc^T, one wave per 16x16 tile
__global__ void __launch_bounds__(128) k_cproj(const float* __restrict__ ptsT,
                                               const float* __restrict__ Wc,
                                               float* __restrict__ cproj) {
    const int wave = threadIdx.x >> 5;
    const int l    = threadIdx.x & 31;
    const int tile = blockIdx.x * 4 + wave;     // 8192 tiles
    const int mt = tile >> 3, ot = tile & 7;
    const int r0 = mt << 4, o0 = ot << 4;
    const int m = l & 15, hi = l >> 4, koff = hi * 2;

    const float* arow = ptsT + ((size_t)(r0 + m)) * DIN;
    const float* wrow = Wc + (o0 + m) * DIN;    // B[k][n] = Wc[o0+n][k]
    v2f a[16], w[16];
#pragma unroll
    for (int c = 0; c < 16; ++c) {
        a[c].x = arow[4 * c + koff]; a[c].y = arow[4 * c + koff + 1];
        w[c].x = wrow[4 * c + koff]; w[c].y = wrow[4 * c + koff + 1];
    }
    v8f acc = {};
#pragma unroll
    for (int c = 0; c < 16; ++c)
        acc = __builtin_amdgcn_wmma_f32_16x16x4_f32(false, a[c], false, w[c],
                                                    (short)0, acc, false, false);
#pragma unroll
    for (int v = 0; v < 8; ++v)
        cproj[((size_t)(r0 + v + 8 * hi)) * DOUT + o0 + m] = acc[v];
}

// ---------------------------------------------------------------- main gather+GEMM
// y[b,o,n,k] = nbr.Wn^T + cproj[b,n,o] + dist*S[o]   (raw, pre-BN) into d_out
__global__ void __launch_bounds__(128) k_main(const float* __restrict__ ptsT,
                                              const float* __restrict__ Wn,
                                              const float* __restrict__ Sv,
                                              const float* __restrict__ cproj,
                                              const int* __restrict__ knn_idx,
                                              const float* __restrict__ knn_dist,
                                              float* __restrict__ y) {
    const int wave = threadIdx.x >> 5;
    const int l    = threadIdx.x & 31;
    const int tile = blockIdx.x * 4 + wave;     // 163840 tiles
    const int mt = tile >> 3, ot = tile & 7;
    const int r0 = mt << 4, o0 = ot << 4;
    const int m = l & 15, hi = l >> 4, koff = hi * 2;

    const int row = r0 + m;                     // row in (b, n, k) space
    const int bn  = row / KNN;
    const int kk  = row - bn * KNN;
    const int j   = knn_idx[bn * KNN + kk];
    const int b   = bn >> 12;
    const float dist = knn_dist[bn * KNN + kk];

    const float* arow = ptsT + ((size_t)((b << 12) + j)) * DIN;
    const float* wrow = Wn + (o0 + m) * DIN;
    v2f a[16], w[16];
#pragma unroll
    for (int c = 0; c < 16; ++c) {
        a[c].x = arow[4 * c + koff]; a[c].y = arow[4 * c + koff + 1];
        w[c].x = wrow[4 * c + koff]; w[c].y = wrow[4 * c + koff + 1];
    }
    v8f acc = {};
#pragma unroll
    for (int c = 0; c < 16; ++c)
        acc = __builtin_amdgcn_wmma_f32_16x16x4_f32(false, a[c], false, w[c],
                                                    (short)0, acc, false, false);

    const float S_o = Sv[o0 + m];
#pragma unroll
    for (int v = 0; v < 8; ++v) {
        const int rr  = r0 + v + 8 * hi;
        const int bnv = rr / KNN;
        const int kv  = rr - bnv * KNN;
        const int bv  = bnv >> 12;
        const int nv  = bnv & (NPTS - 1);
        const float dv = __shfl(dist, v + 8 * hi, 32);
        const float cp = cproj[(size_t)bnv * DOUT + o0 + m];
        const float val = acc[v] + cp + dv * S_o;
        y[((size_t)((bv * DOUT + o0 + m) * NPTS + nv)) * KNN + kv] = val;
    }
}

// ---------------------------------------------------------------- BN stats
__global__ void k_stats(const float* __restrict__ y, float* __restrict__ stats) {
    const int o = blockIdx.x;   // 128
    const int s = blockIdx.y;   // 16 slices: 4 batches x 4 chunks
    const int b = s >> 2, chunk = s & 3;
    const size_t base = ((size_t)(b * DOUT + o)) * NK + (size_t)chunk * (NK / 4);
    float sum = 0.f, sq = 0.f;
    for (int t = threadIdx.x; t < NK / 4; t += 256) {
        float v = y[base + t]; sum += v; sq += v * v;
    }
    __shared__ float rs[256], rq[256];
    rs[threadIdx.x] = sum; rq[threadIdx.x] = sq;
    __syncthreads();
    for (int st = 128; st > 0; st >>= 1) {
        if (threadIdx.x < st) { rs[threadIdx.x] += rs[threadIdx.x + st];
                                rq[threadIdx.x] += rq[threadIdx.x + st]; }
        __syncthreads();
    }
    if (threadIdx.x == 0) {
        atomicAdd(&stats[o], rs[0]);
        atomicAdd(&stats[DOUT + o], rq[0]);
    }
}

// ---------------------------------------------------------------- BN + LeakyReLU
__global__ void k_norm(float* __restrict__ y, const float* __restrict__ stats,
                       const float* __restrict__ gamma, const float* __restrict__ beta) {
    const float invM = 1.0f / ((float)B_ * NK);
    const size_t total = (size_t)B_ * DOUT * NK;
    for (size_t i = (size_t)blockIdx.x * blockDim.x + threadIdx.x; i < total;
         i += (size_t)gridDim.x * blockDim.x) {
        const int o = (int)((i / NK) & (DOUT - 1));
        const float mean = stats[o] * invM;
        const float var  = stats[DOUT + o] * invM - mean * mean;
        const float istd = rsqrtf(var + EPSBN);
        float v = (y[i] - mean) * istd * gamma[o] + beta[o];
        y[i] = v >= 0.f ? v : SLOPE * v;
    }
}

// ---------------------------------------------------------------- launch
extern "C" void kernel_launch(void* const* d_in, const int* in_sizes, int n_in,
                              void* d_out, int out_size, void* d_ws, size_t ws_size,
                              hipStream_t stream) {
    const float* features = (const float*)d_in[0];   // (4, 64, 4096, 1)
    const float* conv_w   = (const float*)d_in[1];   // (128, 256)
    const float* gamma    = (const float*)d_in[2];   // (128,)
    const float* beta     = (const float*)d_in[3];   // (128,)
    float* y = (float*)d_out;                        // (4, 128, 4096, 20)

    char* ws = (char*)d_ws;
    float* ptsT  = (float*)(ws);                                  // 4,194,304 B
    float* Wc    = (float*)(ws + 4194304);                        //    32,768 B
    float* Wn    = (float*)(ws + 4227072);                        //    32,768 B
    float* Sv    = (float*)(ws + 4259840);                        //     1,024 B
    float* stats = (float*)(ws + 4260864);                        //     1,024 B
    int*   kidx  = (int*)  (ws + 4261888);                        // 1,310,720 B
    float* kdst  = (float*)(ws + 4261888 + 1310720);              // 1,310,720 B
    float* cproj = (float*)(ws + 4261888 + 2 * 1310720);          // 8,388,608 B

    k_prep     <<<128, 64, 0, stream>>>(conv_w, Wc, Wn, Sv, stats);
    k_transpose<<<dim3(NPTS / 32, DIN / 32, B_), dim3(32, 8), 0, stream>>>(features, ptsT);
    k_knn      <<<B_ * (NPTS / 16), 32, 0, stream>>>(ptsT, kidx, kdst);
    k_cproj    <<<(B_ * NPTS / 16) * (DOUT / 16) / 4, 128, 0, stream>>>(ptsT, Wc, cproj);
    k_main     <<<(B_ * NK / 16) * (DOUT / 16) / 4, 128, 0, stream>>>(ptsT, Wn, Sv, cproj,
                                                                      kidx, kdst, y);
    k_stats    <<<dim3(DOUT, 16), 256, 0, stream>>>(y, stats);
    k_norm     <<<4096, 256, 0, stream>>>(y, stats, gamma, beta);
}